// MambaSSM_22952305230043
// MI455X (gfx1250) — compile-verified
//
#include <hip/hip_runtime.h>
#include <hip/hip_bf16.h>

typedef float v2f __attribute__((ext_vector_type(2)));
typedef float v4f __attribute__((ext_vector_type(4)));
typedef float v8f __attribute__((ext_vector_type(8)));
typedef int   async_v4i __attribute__((vector_size(16)));   // matches builtin param

#define DIM      768
#define D_STATE  16
#define D_CONV   4
#define D_INNER  1536
#define SEQ      1024
#define NLAYERS  4
#define XPROJ_N  (D_INNER + 2 * D_STATE)   // 1568
#define VOCAB    50257

#define LDS_AS  __attribute__((address_space(3)))
#define GLB_AS  __attribute__((address_space(1)))

#if defined(__has_builtin)
#if __has_builtin(__builtin_amdgcn_global_load_async_to_lds_b128) && \
    __has_builtin(__builtin_amdgcn_s_wait_asynccnt)
#define USE_ASYNC_LDS 1
#endif
#endif

// ---------------------------------------------------------------------------
// Tiled fp32 WMMA GEMM:  C[M,N] = A[M,K] (row-major, stride lda) * W[N,K]^T
// Optional residual add (res != nullptr -> C = res + A*W^T).
// Block: 256 threads (8 waves, 2x4 arrangement). Block tile 128(M) x 128(N).
// Wave tile: 64(M) x 32(N) = 4 x 2 accumulators; every A-fragment feeds two
// WMMAs (halves LDS traffic per FLOP), every B-fragment feeds four.
// A staged into LDS; when available, via GLOBAL_LOAD_ASYNC_TO_LDS_B128
// (ASYNCcnt, double-buffered), else via VGPR round-trip.
// ---------------------------------------------------------------------------
__global__ __launch_bounds__(256) void gemm_wmma_f32(
    const float* __restrict__ A, const float* __restrict__ W,
    const float* __restrict__ res, float* __restrict__ C,
    int M, int N, int K, int lda)
{
    __shared__ float As[2][128 * 36];      // 128 rows x 32 K, padded stride 36

    const int lane = threadIdx.x & 31;
    const int wave = threadIdx.x >> 5;
    const int wm   = wave >> 2;            // 0..1  (M position)
    const int wn   = wave & 3;             // 0..3  (N position)
    const int m0   = blockIdx.y * 128;

    const int ncol0   = blockIdx.x * 128 + wn * 32 + (lane & 15);
    const int ncol1   = ncol0 + 16;
    const int nclamp0 = (ncol0 < N) ? ncol0 : (N - 1);
    const int nclamp1 = (ncol1 < N) ? ncol1 : (N - 1);
    const int koff    = (lane < 16) ? 0 : 2;       // K-pair per half-wave

    const v8f zero = {0.f, 0.f, 0.f, 0.f, 0.f, 0.f, 0.f, 0.f};
    v8f acc[4][2] = {{zero, zero}, {zero, zero}, {zero, zero}, {zero, zero}};

    const float* __restrict__ wrow0 = W + (size_t)nclamp0 * K;
    const float* __restrict__ wrow1 = W + (size_t)nclamp1 * K;

    // Staging geometry: 128 rows x 32 floats = 1024 float4; 4 per thread.
#if USE_ASYNC_LDS
    {   // prologue: stream chunk 0 into buffer 0
#pragma unroll
        for (int r = 0; r < 4; r++) {
            int idx = threadIdx.x + r * 256;
            int row = idx >> 3, kq = (idx & 7) * 4;
            const float* src = A + (size_t)(m0 + row) * lda + kq;
            float* dst = &As[0][row * 36 + kq];
            __builtin_amdgcn_global_load_async_to_lds_b128(
                (GLB_AS async_v4i*)(uintptr_t)src,
                (LDS_AS async_v4i*)(uint32_t)(uintptr_t)dst, 0, 0);
        }
    }
#endif

    for (int kc = 0; kc < K; kc += 32) {
#if USE_ASYNC_LDS
        const int cur  = (kc >> 5) & 1;
        const bool more = (kc + 32 < K);
        if (more) {
#pragma unroll
            for (int r = 0; r < 4; r++) {
                int idx = threadIdx.x + r * 256;
                int row = idx >> 3, kq = (idx & 7) * 4;
                const float* src = A + (size_t)(m0 + row) * lda + kc + 32 + kq;
                float* dst = &As[cur ^ 1][row * 36 + kq];
                __builtin_amdgcn_global_load_async_to_lds_b128(
                    (GLB_AS async_v4i*)(uintptr_t)src,
                    (LDS_AS async_v4i*)(uint32_t)(uintptr_t)dst, 0, 0);
            }
            __builtin_amdgcn_s_wait_asynccnt(4);   // this chunk landed
        } else {
            __builtin_amdgcn_s_wait_asynccnt(0);
        }
        __syncthreads();                           // all waves' data visible
        const float* __restrict__ Ab = &As[cur][0];
#else
        __syncthreads();
#pragma unroll
        for (int r = 0; r < 4; r++) {
            int idx = threadIdx.x + r * 256;
            int row = idx >> 3, kq = (idx & 7) * 4;
            *(v4f*)&As[0][row * 36 + kq] =
                *(const v4f*)(A + (size_t)(m0 + row) * lda + kc + kq);
        }
        __syncthreads();
        const float* __restrict__ Ab = &As[0][0];
#endif

        if (kc + 32 < K) {
            __builtin_prefetch(wrow0 + kc + 32, 0, 1);   // global_prefetch_b8
            __builtin_prefetch(wrow1 + kc + 32, 0, 1);
        }

#pragma unroll
        for (int ks = 0; ks < 32; ks += 4) {
            v2f b0 = *(const v2f*)&wrow0[kc + ks + koff];
            v2f b1 = *(const v2f*)&wrow1[kc + ks + koff];
#pragma unroll
            for (int mi = 0; mi < 4; mi++) {
                v2f a = *(const v2f*)&Ab[(wm * 64 + mi * 16 + (lane & 15)) * 36
                                         + ks + koff];
                acc[mi][0] = __builtin_amdgcn_wmma_f32_16x16x4_f32(
                    false, a, false, b0, (short)0, acc[mi][0], false, false);
                acc[mi][1] = __builtin_amdgcn_wmma_f32_16x16x4_f32(
                    false, a, false, b1, (short)0, acc[mi][1], false, false);
            }
        }
#if USE_ASYNC_LDS
        __syncthreads();   // finish reading As[cur] before it is refilled
#endif
    }

    const int rbase = ((lane >> 4) << 3);          // 0 or 8
#pragma unroll
    for (int mi = 0; mi < 4; mi++) {
#pragma unroll
        for (int e = 0; e < 8; e++) {
            int row = m0 + wm * 64 + mi * 16 + rbase + e;
            if (ncol0 < N) {
                size_t o = (size_t)row * N + ncol0;
                float v = acc[mi][0][e];
                if (res) v += res[o];
                C[o] = v;
            }
            if (ncol1 < N) {
                size_t o = (size_t)row * N + ncol1;
                float v = acc[mi][1][e];
                if (res) v += res[o];
                C[o] = v;
            }
        }
    }
}

// ---------------------------------------------------------------------------
// Embedding gather: x[s, :] = embedding[tokens[s], :]
// ---------------------------------------------------------------------------
__global__ __launch_bounds__(256) void embed_kernel(
    const int* __restrict__ tok, const float* __restrict__ emb,
    float* __restrict__ x)
{
    int s = blockIdx.x, t = threadIdx.x;
    const float* e = emb + (size_t)tok[s] * DIM;
    float* o = x + (size_t)s * DIM;
    o[t]       = e[t];
    o[t + 256] = e[t + 256];
    o[t + 512] = e[t + 512];
}

// ---------------------------------------------------------------------------
// RMSNorm over DIM=768, one block per row
// ---------------------------------------------------------------------------
__global__ __launch_bounds__(256) void rmsnorm_kernel(
    const float* __restrict__ x, const float* __restrict__ w,
    float* __restrict__ out)
{
    __shared__ float red[256];
    int s = blockIdx.x, t = threadIdx.x;
    const float* row = x + (size_t)s * DIM;
    float v0 = row[t], v1 = row[t + 256], v2 = row[t + 512];
    red[t] = v0 * v0 + v1 * v1 + v2 * v2;
    __syncthreads();
    for (int off = 128; off > 0; off >>= 1) {
        if (t < off) red[t] += red[t + off];
        __syncthreads();
    }
    float rs = rsqrtf(red[0] * (1.0f / DIM) + 1e-6f);
    float* o = out + (size_t)s * DIM;
    o[t]       = w[t]       * v0 * rs;
    o[t + 256] = w[t + 256] * v1 * rs;
    o[t + 512] = w[t + 512] * v2 * rs;
}

// ---------------------------------------------------------------------------
// Causal depthwise conv (D_CONV=4) + bias + SiLU over first half of xz
// ---------------------------------------------------------------------------
__global__ __launch_bounds__(256) void conv_silu_kernel(
    const float* __restrict__ xz, const float* __restrict__ cw,
    const float* __restrict__ cb, float* __restrict__ xc)
{
    int idx = blockIdx.x * 256 + threadIdx.x;           // s * D_INNER + d
    int s = idx / D_INNER, d = idx - s * D_INNER;
    float acc = cb[d];
#pragma unroll
    for (int j = 0; j < D_CONV; j++) {
        int sp = s - (D_CONV - 1) + j;
        if (sp >= 0)
            acc += cw[d * D_CONV + j] * xz[(size_t)sp * (2 * D_INNER) + d];
    }
    xc[idx] = acc / (1.f + __expf(-acc));               // SiLU
}

// ---------------------------------------------------------------------------
// Selective-scan recurrence, fused softplus(dt)+bias, D-skip and SiLU(z) gate
// ---------------------------------------------------------------------------
__global__ __launch_bounds__(256) void scan_kernel(
    const float* __restrict__ xc,   const float* __restrict__ xp,
    const float* __restrict__ dtr,  const float* __restrict__ dt_b,
    const float* __restrict__ A_log,const float* __restrict__ Dp,
    const float* __restrict__ xz,   float* __restrict__ y)
{
    int d = blockIdx.x * 256 + threadIdx.x;             // 0..D_INNER-1
    float A[D_STATE], h[D_STATE];
#pragma unroll
    for (int n = 0; n < D_STATE; n++) {
        A[n] = -__expf(A_log[(size_t)d * D_STATE + n]);
        h[n] = 0.f;
    }
    const float Dd = Dp[d], dtb = dt_b[d];

    for (int s = 0; s < SEQ; s++) {
        float dtv = dtr[(size_t)s * D_INNER + d] + dtb;
        float dt  = (dtv > 20.f) ? dtv : __logf(1.f + __expf(dtv));
        float xv  = xc[(size_t)s * D_INNER + d];
        float zv  = xz[(size_t)s * (2 * D_INNER) + D_INNER + d];
        const float* Bm = xp + (size_t)s * XPROJ_N + D_INNER;
        const float* Cm = Bm + D_STATE;
        float dtx = dt * xv;
        float acc = 0.f;
#pragma unroll
        for (int n = 0; n < D_STATE; n++) {
            h[n] = h[n] * __expf(dt * A[n]) + dtx * Bm[n];
            acc += h[n] * Cm[n];
        }
        float sz = zv / (1.f + __expf(-zv));
        y[(size_t)s * D_INNER + d] = (acc + Dd * xv) * sz;
    }
}

// ---------------------------------------------------------------------------
extern "C" void kernel_launch(void* const* d_in, const int* in_sizes, int n_in,
                              void* d_out, int out_size, void* d_ws, size_t ws_size,
                              hipStream_t stream)
{
    (void)in_sizes; (void)n_in; (void)out_size; (void)ws_size;

    const int*   tokens     = (const int*)  d_in[0];
    const float* embedding  = (const float*)d_in[1];
    const float* norm_w     = (const float*)d_in[2];
    const float* in_proj_w  = (const float*)d_in[3];
    const float* conv_w     = (const float*)d_in[4];
    const float* conv_b     = (const float*)d_in[5];
    const float* x_proj_w   = (const float*)d_in[6];
    const float* dt_w       = (const float*)d_in[7];
    const float* dt_b       = (const float*)d_in[8];
    const float* A_log      = (const float*)d_in[9];
    const float* Dparam     = (const float*)d_in[10];
    const float* out_proj_w = (const float*)d_in[11];
    const float* normf_w    = (const float*)d_in[12];
    const float* lm_head_w  = (const float*)d_in[13];
    float* logits           = (float*)d_out;

    float* ws  = (float*)d_ws;
    float* x0  = ws;                                   // 1024*768
    float* x1  = ws + 786432;                          // 1024*768
    float* xn  = ws + 1572864;                         // 1024*768
    float* xz  = ws + 2359296;                         // 1024*3072
    float* xc  = ws + 5505024;                         // 1024*1536
    float* xp  = ws + 7077888;                         // 1024*1568
    float* dtr = ws + 8683520;                         // 1024*1536
    float* yb  = ws + 10256384;                        // 1024*1536

    embed_kernel<<<SEQ, 256, 0, stream>>>(tokens, embedding, x0);

    float* cur = x0;
    float* nxt = x1;
    const int MB = SEQ / 128;                          // 8 M-blocks

    for (int l = 0; l < NLAYERS; l++) {
        const float* nw  = norm_w     + (size_t)l * DIM;
        const float* ipw = in_proj_w  + (size_t)l * 2 * D_INNER * DIM;
        const float* cw  = conv_w     + (size_t)l * D_INNER * D_CONV;
        const float* cb  = conv_b     + (size_t)l * D_INNER;
        const float* xpw = x_proj_w   + (size_t)l * XPROJ_N * D_INNER;
        const float* dtw = dt_w       + (size_t)l * D_INNER * D_INNER;
        const float* dtb = dt_b       + (size_t)l * D_INNER;
        const float* al  = A_log      + (size_t)l * D_INNER * D_STATE;
        const float* dp  = Dparam     + (size_t)l * D_INNER;
        const float* opw = out_proj_w + (size_t)l * DIM * D_INNER;

        rmsnorm_kernel<<<SEQ, 256, 0, stream>>>(cur, nw, xn);

        gemm_wmma_f32<<<dim3((2 * D_INNER + 127) / 128, MB), 256, 0, stream>>>(
            xn, ipw, nullptr, xz, SEQ, 2 * D_INNER, DIM, DIM);

        conv_silu_kernel<<<(SEQ * D_INNER) / 256, 256, 0, stream>>>(xz, cw, cb, xc);

        gemm_wmma_f32<<<dim3((XPROJ_N + 127) / 128, MB), 256, 0, stream>>>(
            xc, xpw, nullptr, xp, SEQ, XPROJ_N, D_INNER, D_INNER);

        gemm_wmma_f32<<<dim3((D_INNER + 127) / 128, MB), 256, 0, stream>>>(
            xp, dtw, nullptr, dtr, SEQ, D_INNER, D_INNER, XPROJ_N);

        scan_kernel<<<D_INNER / 256, 256, 0, stream>>>(xc, xp, dtr, dtb, al, dp, xz, yb);

        gemm_wmma_f32<<<dim3((DIM + 127) / 128, MB), 256, 0, stream>>>(
            yb, opw, cur, nxt, SEQ, DIM, D_INNER, D_INNER);

        float* t = cur; cur = nxt; nxt = t;
    }

    rmsnorm_kernel<<<SEQ, 256, 0, stream>>>(cur, normf_w, xn);

    gemm_wmma_f32<<<dim3((VOCAB + 127) / 128, MB), 256, 0, stream>>>(
        xn, lm_head_w, nullptr, logits, SEQ, VOCAB, DIM, DIM);
}